// PyG_GCN_52158082842625
// MI455X (gfx1250) — compile-verified
//
#include <hip/hip_runtime.h>
#include <hip/hip_bf16.h>

typedef __attribute__((ext_vector_type(2))) float v2f;
typedef __attribute__((ext_vector_type(8))) float v8f;

#define FEAT 128
#define NGRAPH 256
#define NCLS 10

// ---------------- degree / norm ----------------
__global__ void k_deg_init(float* deg, int n) {
    int i = blockIdx.x * blockDim.x + threadIdx.x;
    if (i < n) deg[i] = 1.0f;  // self-loop contributes 1
}

__global__ void k_deg_edges(const int* __restrict__ ei, float* deg, int nEdges) {
    int e = blockIdx.x * blockDim.x + threadIdx.x;
    if (e < nEdges) atomicAdd(&deg[ei[nEdges + e]], 1.0f);  // dst row
}

__global__ void k_make_dis(const float* __restrict__ deg, float* dis, int n) {
    int i = blockIdx.x * blockDim.x + threadIdx.x;
    if (i < n) dis[i] = rsqrtf(deg[i]);  // deg >= 1 always (self-loops)
}

// ---------------- fp32 WMMA GEMM: out[M,128] = act[M,128] @ W[128,128] ----------------
// block = 256 threads = 8 waves. Each wave owns TWO 16x16 output tiles (rows r..r+15 and
// r+16..r+31) sharing one B fragment per K-step; block covers 32 rows x 128 cols.
// RELU is a compile-time flag so layer-1 gets a clean loop and layers 2/3 get a bare v_max.
template <int RELU>
__global__ void k_gemm_wmma(const float* __restrict__ act, const float* __restrict__ W,
                            float* __restrict__ out) {
    const int lane    = threadIdx.x & 31;
    const int wave    = threadIdx.x >> 5;
    const int rowBase = blockIdx.x * 32;
    const int colBase = wave * 16;
    const int half    = lane >> 4;   // 0: lanes 0-15, 1: lanes 16-31
    const int lmod    = lane & 15;
    const int col     = colBase + lmod;          // B-frag col for this lane

    const float* arow0 = act + (size_t)(rowBase + lmod) * FEAT;        // tile 0 row
    const float* arow1 = act + (size_t)(rowBase + 16 + lmod) * FEAT;   // tile 1 row
    v8f c0 = {};
    v8f c1 = {};
    for (int k0 = 0; k0 < FEAT; k0 += 4) {
        const int ka = k0 + 2 * half;
        v2f a0, a1, b;
        a0.x = arow0[ka];
        a0.y = arow0[ka + 1];
        a1.x = arow1[ka];
        a1.y = arow1[ka + 1];
        if (RELU) {
            a0.x = fmaxf(a0.x, 0.0f); a0.y = fmaxf(a0.y, 0.0f);
            a1.x = fmaxf(a1.x, 0.0f); a1.y = fmaxf(a1.y, 0.0f);
        }
        b.x = W[(size_t)ka * FEAT + col];
        b.y = W[(size_t)(ka + 1) * FEAT + col];
        c0 = __builtin_amdgcn_wmma_f32_16x16x4_f32(false, a0, false, b,
                                                   (short)0, c0, false, false);
        c1 = __builtin_amdgcn_wmma_f32_16x16x4_f32(false, a1, false, b,
                                                   (short)0, c1, false, false);
    }
    // C/D layout: VGPR v, lanes 0-15 -> row v, lanes 16-31 -> row v+8; col = lane%16
#pragma unroll
    for (int v = 0; v < 8; ++v) {
        out[(size_t)(rowBase + v + 8 * half) * FEAT + col]      = c0[v];
        out[(size_t)(rowBase + 16 + v + 8 * half) * FEAT + col] = c1[v];
    }
}

// ---------------- agg init: self-loop message + bias ----------------
__global__ void k_selfloop_bias(const float* __restrict__ h, const float* __restrict__ dis,
                                const float* __restrict__ bias, float* __restrict__ agg, int n) {
    int idx = blockIdx.x * blockDim.x + threadIdx.x;
    if (idx >= n * FEAT) return;
    int i = idx >> 7;
    int cIdx = idx & (FEAT - 1);
    float di = dis[i];
    agg[idx] = h[idx] * di * di + bias[cIdx];
}

// ---------------- edge scatter: one wave32 per edge, float4 per lane ----------------
__global__ void k_scatter_edges(const float* __restrict__ h, const int* __restrict__ ei,
                                const float* __restrict__ dis, float* __restrict__ agg,
                                int nEdges) {
    int e    = blockIdx.x * (blockDim.x >> 5) + (threadIdx.x >> 5);
    int lane = threadIdx.x & 31;
    if (e >= nEdges) return;
    int s = ei[e];
    int d = ei[nEdges + e];
    float w = dis[s] * dis[d];
    const float4* hv = (const float4*)(h + (size_t)s * FEAT);
    float4 v = hv[lane];                 // 32 lanes x float4 = full 128-float row
    float* ag = agg + (size_t)d * FEAT + lane * 4;
    atomicAdd(ag + 0, v.x * w);
    atomicAdd(ag + 1, v.y * w);
    atomicAdd(ag + 2, v.z * w);
    atomicAdd(ag + 3, v.w * w);
}

// ---------------- pooling ----------------
__global__ void k_pool_zero(float* sums, float* cnts) {
    int i = blockIdx.x * blockDim.x + threadIdx.x;
    if (i < NGRAPH * FEAT) sums[i] = 0.0f;
    if (i < NGRAPH) cnts[i] = 0.0f;
}

__global__ void k_pool_accum(const float* __restrict__ h, const int* __restrict__ batch,
                             float* sums, float* cnts, int n) {
    int i    = blockIdx.x * (blockDim.x >> 5) + (threadIdx.x >> 5);
    int lane = threadIdx.x & 31;
    if (i >= n) return;
    int g = batch[i];
    if (lane == 0) atomicAdd(&cnts[g], 1.0f);
    const float4* hv = (const float4*)(h + (size_t)i * FEAT);
    float4 v = hv[lane];
    float* sg = sums + (size_t)g * FEAT + lane * 4;
    atomicAdd(sg + 0, fmaxf(v.x, 0.0f));   // ReLU of layer-3 output folded in
    atomicAdd(sg + 1, fmaxf(v.y, 0.0f));
    atomicAdd(sg + 2, fmaxf(v.z, 0.0f));
    atomicAdd(sg + 3, fmaxf(v.w, 0.0f));
}

// ---------------- head MLP ----------------
__global__ void k_mlp1(const float* __restrict__ sums, const float* __restrict__ cnts,
                       const float* __restrict__ W, const float* __restrict__ b,
                       float* __restrict__ g1) {
    int g = blockIdx.x;       // 256
    int j = threadIdx.x;      // 128
    float inv = 1.0f / fmaxf(cnts[g], 1.0f);
    float acc = 0.0f;
    for (int cc = 0; cc < FEAT; ++cc) acc += sums[g * FEAT + cc] * W[cc * FEAT + j];
    g1[g * FEAT + j] = fmaxf(acc * inv + b[j], 0.0f);   // mean is linear: (sum@W)/cnt
}

__global__ void k_mlp2(const float* __restrict__ g1, const float* __restrict__ W,
                       const float* __restrict__ b, float* __restrict__ out) {
    int g = blockIdx.x;       // 256
    int k = threadIdx.x;      // 32, guard to 10
    if (k < NCLS) {
        float acc = b[k];
        for (int j = 0; j < FEAT; ++j) acc += g1[g * FEAT + j] * W[j * NCLS + k];
        out[g * NCLS + k] = acc;
    }
}

extern "C" void kernel_launch(void* const* d_in, const int* in_sizes, int n_in,
                              void* d_out, int out_size, void* d_ws, size_t ws_size,
                              hipStream_t stream) {
    (void)n_in; (void)out_size; (void)ws_size;
    const float* x      = (const float*)d_in[0];
    const int*   ei     = (const int*)d_in[1];
    const int*   batch  = (const int*)d_in[2];
    const float* W1     = (const float*)d_in[3];
    const float* b1     = (const float*)d_in[4];
    const float* W2     = (const float*)d_in[5];
    const float* b2     = (const float*)d_in[6];
    const float* W3     = (const float*)d_in[7];
    const float* b3     = (const float*)d_in[8];
    const float* lin1w  = (const float*)d_in[9];
    const float* lin1b  = (const float*)d_in[10];
    const float* lin2w  = (const float*)d_in[11];
    const float* lin2b  = (const float*)d_in[12];
    float* out = (float*)d_out;

    const int nNodes = in_sizes[0] / FEAT;      // 100000
    const int nEdges = in_sizes[1] / 2;         // 3200000

    // workspace layout (floats); nNodes % 4 == 0 keeps float4 alignment
    float* ws   = (float*)d_ws;
    size_t off  = 0;
    float* deg  = ws + off; off += (size_t)nNodes;
    float* dis  = ws + off; off += (size_t)nNodes;
    float* bufH = ws + off; off += (size_t)nNodes * FEAT;   // GEMM output
    float* bufA = ws + off; off += (size_t)nNodes * FEAT;   // aggregation output / next act
    float* sums = ws + off; off += (size_t)NGRAPH * FEAT;
    float* cnts = ws + off; off += (size_t)NGRAPH;
    float* g1   = ws + off; off += (size_t)NGRAPH * FEAT;

    const int B = 256;
    // degrees / symmetric norm
    k_deg_init <<<(nNodes + B - 1) / B, B, 0, stream>>>(deg, nNodes);
    k_deg_edges<<<(nEdges + B - 1) / B, B, 0, stream>>>(ei, deg, nEdges);
    k_make_dis <<<(nNodes + B - 1) / B, B, 0, stream>>>(deg, dis, nNodes);

    const int gemmGrid = nNodes / 32;                       // 3125, exact
    const int nfGrid   = (nNodes * FEAT + B - 1) / B;
    const int edgeGrid = (nEdges + 7) / 8;                  // 8 edges (waves) per block

    const float* Ws[3] = {W1, W2, W3};
    const float* bs[3] = {b1, b2, b3};
    for (int layer = 0; layer < 3; ++layer) {
        const float* act = (layer == 0) ? x : bufA;
        if (layer == 0)
            k_gemm_wmma<0><<<gemmGrid, B, 0, stream>>>(act, Ws[layer], bufH);
        else
            k_gemm_wmma<1><<<gemmGrid, B, 0, stream>>>(act, Ws[layer], bufH);
        k_selfloop_bias<<<nfGrid, B, 0, stream>>>(bufH, dis, bs[layer], bufA, nNodes);
        k_scatter_edges<<<edgeGrid, B, 0, stream>>>(bufH, ei, dis, bufA, nEdges);
    }

    // global mean pool (ReLU folded into accumulation)
    k_pool_zero <<<(NGRAPH * FEAT + B - 1) / B, B, 0, stream>>>(sums, cnts);
    k_pool_accum<<<(nNodes + 7) / 8, B, 0, stream>>>(bufA, batch, sums, cnts, nNodes);

    // head
    k_mlp1<<<NGRAPH, FEAT, 0, stream>>>(sums, cnts, lin1w, lin1b, g1);
    k_mlp2<<<NGRAPH, 32, 0, stream>>>(g1, lin2w, lin2b, out);
}